// DeDoDeDetector_42863773614540
// MI455X (gfx1250) — compile-verified
//
#include <hip/hip_runtime.h>
#include <stdint.h>

typedef float v2f __attribute__((ext_vector_type(2)));
typedef float v8f __attribute__((ext_vector_type(8)));

#define BB 16
#define HH 784
#define WW 784
#define HWV (HH * WW)
#define NK 8192
#define CPAD 25
#define KTOT 68          // 66 needed (16 outputs + 50 overlap), padded to 17*4
#define TPB 49           // 784/16 tiles per dimension
#define TILES_B (TPB * TPB)
#define NEGINF (-3.4028234663852886e38f)

struct TopkState {
    unsigned long long prefix;
    unsigned int kr;
    unsigned int cnt;
};

__device__ __forceinline__ float wcov(int d) {
    // w1d[d] = exp(-linspace(-2,2,51)[d]^2), zero outside [0,50]
    if (d < 0 || d > 50) return 0.0f;
    float t = -2.0f + 0.08f * (float)d;
    return expf(-t * t);
}

__device__ __forceinline__ unsigned long long makeKey(float v, int idx) {
    // v >= 0 always -> float bits are monotonic; ~idx => ties pick lowest index
    unsigned int fb = __float_as_uint(v);
    return ((unsigned long long)fb << 32) | (unsigned int)(~idx);
}

// ---------------- softmax reductions ----------------

__global__ void reduceMaxK(const float* __restrict__ sc, float* __restrict__ partial) {
    __shared__ float red[256];
    int b = blockIdx.y;
    const float* p = sc + (size_t)b * HWV;
    float m = NEGINF;
    for (int i = blockIdx.x * 256 + threadIdx.x; i < HWV; i += 256 * gridDim.x)
        m = fmaxf(m, p[i]);
    red[threadIdx.x] = m;
    __syncthreads();
    for (int s = 128; s > 0; s >>= 1) {
        if (threadIdx.x < s) red[threadIdx.x] = fmaxf(red[threadIdx.x], red[threadIdx.x + s]);
        __syncthreads();
    }
    if (threadIdx.x == 0) partial[b * gridDim.x + blockIdx.x] = red[0];
}

__global__ void finalMaxK(const float* __restrict__ partial, float* __restrict__ maxv) {
    __shared__ float red[256];
    int b = blockIdx.x;
    red[threadIdx.x] = partial[b * 256 + threadIdx.x];
    __syncthreads();
    for (int s = 128; s > 0; s >>= 1) {
        if (threadIdx.x < s) red[threadIdx.x] = fmaxf(red[threadIdx.x], red[threadIdx.x + s]);
        __syncthreads();
    }
    if (threadIdx.x == 0) maxv[b] = red[0];
}

__global__ void reduceSumK(const float* __restrict__ sc, const float* __restrict__ maxv,
                           float* __restrict__ partial) {
    __shared__ float red[256];
    int b = blockIdx.y;
    const float* p = sc + (size_t)b * HWV;
    float mb = maxv[b];
    float s = 0.0f;
    for (int i = blockIdx.x * 256 + threadIdx.x; i < HWV; i += 256 * gridDim.x)
        s += expf(p[i] - mb);
    red[threadIdx.x] = s;
    __syncthreads();
    for (int st = 128; st > 0; st >>= 1) {
        if (threadIdx.x < st) red[threadIdx.x] += red[threadIdx.x + st];
        __syncthreads();
    }
    if (threadIdx.x == 0) partial[b * gridDim.x + blockIdx.x] = red[0];
}

__global__ void finalSumK(const float* __restrict__ partial, float* __restrict__ invs) {
    __shared__ float red[256];
    int b = blockIdx.x;
    red[threadIdx.x] = partial[b * 256 + threadIdx.x];
    __syncthreads();
    for (int s = 128; s > 0; s >>= 1) {
        if (threadIdx.x < s) red[threadIdx.x] += red[threadIdx.x + s];
        __syncthreads();
    }
    if (threadIdx.x == 0) invs[b] = 1.0f / red[0];
}

// ---------------- horizontal coverage conv as WMMA Toeplitz GEMM ----------------
// out[r, c0+c] = sum_k x[r, c0+k-25] * W[k,c],  W[k,c] = w1d[k-c]
// 4 waves per block, one 16x16 output tile per wave, 17 chunks of V_WMMA_F32_16X16X4_F32.

__global__ void __launch_bounds__(128) convHK(const float* __restrict__ sc,
                                              const float* __restrict__ maxv,
                                              const float* __restrict__ invs,
                                              float* __restrict__ xh) {
    __shared__ float Wt[KTOT][16];       // Toeplitz weights (shared by all waves)
    __shared__ float Ain[4][16][KTOT];   // per-wave staged input tile

    for (int i = threadIdx.x; i < KTOT * 16; i += blockDim.x) {
        int k = i / 16, n = i % 16;
        Wt[k][n] = wcov(k - n);
    }

    int lane = threadIdx.x & 31;
    int wv = threadIdx.x >> 5;
    int tile = blockIdx.x * 4 + wv;
    int b = tile / TILES_B;
    int t = tile % TILES_B;
    int row0 = (t / TPB) * 16;
    int col0 = (t % TPB) * 16;

    float mxb = maxv[b];
    float is = invs[b];
    const float* sb = sc + (size_t)b * HWV;

    // stage x = (softmax_dense + 1e-6) * 1e4, zero outside image (conv zero-pad)
    for (int r2 = 0; r2 < 16; ++r2) {
        for (int c2 = lane; c2 < KTOT; c2 += 32) {
            int cc = col0 + c2 - CPAD;
            float x = 0.0f;
            if (c2 < 66 && cc >= 0 && cc < WW) {
                float d = expf(sb[(size_t)(row0 + r2) * WW + cc] - mxb) * is;
                x = (d + 1e-6f) * 1e4f;
            }
            Ain[wv][r2][c2] = x;
        }
    }
    __syncthreads();

    int m = lane & 15;     // A row (lanes 0-15 and 16-31 both map M=0..15)
    int kh = lane >> 4;    // K half select
    v8f acc = {};
#pragma unroll
    for (int kc = 0; kc < 17; ++kc) {
        int kb = kc * 4;
        int ka = kb + (kh ? 2 : 0);
        v2f a, bv;
        a.x = Ain[wv][m][ka];
        a.y = Ain[wv][m][ka + 1];
        bv.x = Wt[ka][m];        // B column index == lane&15 as well
        bv.y = Wt[ka + 1][m];
        acc = __builtin_amdgcn_wmma_f32_16x16x4_f32(false, a, false, bv,
                                                    (short)0, acc, false, false);
    }

    float* xb = xh + (size_t)b * HWV;
    int rbase = row0 + (kh ? 8 : 0);
#pragma unroll
    for (int j = 0; j < 8; ++j)
        xb[(size_t)(rbase + j) * WW + col0 + m] = acc[j];
}

// ---------------- vertical conv (weights in A, data in B) + probs ----------------
// ld[r0+r, c] = sum_k W[k,r] * xh[r0+k-25, c];  probs = dense / sqrt(ld + 1e-8)

__global__ void __launch_bounds__(128) convVK(const float* __restrict__ sc,
                                              const float* __restrict__ xh,
                                              const float* __restrict__ maxv,
                                              const float* __restrict__ invs,
                                              float* __restrict__ probs) {
    __shared__ float Wt[KTOT][16];
    __shared__ float Bin[4][KTOT][16];

    for (int i = threadIdx.x; i < KTOT * 16; i += blockDim.x) {
        int k = i / 16, n = i % 16;
        Wt[k][n] = wcov(k - n);
    }

    int lane = threadIdx.x & 31;
    int wv = threadIdx.x >> 5;
    int tile = blockIdx.x * 4 + wv;
    int b = tile / TILES_B;
    int t = tile % TILES_B;
    int row0 = (t / TPB) * 16;
    int col0 = (t % TPB) * 16;

    const float* xb = xh + (size_t)b * HWV;
    for (int i = lane; i < KTOT * 16; i += 32) {
        int k = i / 16, c = i % 16;
        int rr = row0 + k - CPAD;
        float v = 0.0f;
        if (k < 66 && rr >= 0 && rr < HH) v = xb[(size_t)rr * WW + col0 + c];
        Bin[wv][k][c] = v;
    }
    __syncthreads();

    int m = lane & 15;
    int kh = lane >> 4;
    v8f acc = {};
#pragma unroll
    for (int kc = 0; kc < 17; ++kc) {
        int kb = kc * 4;
        int ka = kb + (kh ? 2 : 0);
        v2f a, bv;
        a.x = Wt[ka][m];            // A[m][k] = w1d[k-m]
        a.y = Wt[ka + 1][m];
        bv.x = Bin[wv][ka][m];
        bv.y = Bin[wv][ka + 1][m];
        acc = __builtin_amdgcn_wmma_f32_16x16x4_f32(false, a, false, bv,
                                                    (short)0, acc, false, false);
    }

    float mxb = maxv[b];
    float is = invs[b];
    const float* sb = sc + (size_t)b * HWV;
    float* pb = probs + (size_t)b * HWV;
    int rbase = row0 + (kh ? 8 : 0);
#pragma unroll
    for (int j = 0; j < 8; ++j) {
        size_t off = (size_t)(rbase + j) * WW + col0 + m;
        float dense = expf(sb[off] - mxb) * is;
        pb[off] = dense / sqrtf(acc[j] + 1e-8f);
    }
}

// ---------------- NMS 3x3 + border ----------------

__global__ void nmsK(const float* __restrict__ probs, float* __restrict__ out) {
    int b = blockIdx.y;
    int i = blockIdx.x * 256 + threadIdx.x;
    if (i >= HWV) return;
    int r = i / WW, c = i % WW;
    const float* pb = probs + (size_t)b * HWV;
    float p = pb[i];
    float mx = NEGINF;
#pragma unroll
    for (int dr = -1; dr <= 1; ++dr)
#pragma unroll
        for (int dc = -1; dc <= 1; ++dc) {
            int rr = r + dr, cc = c + dc;
            if (rr >= 0 && rr < HH && cc >= 0 && cc < WW)
                mx = fmaxf(mx, pb[(size_t)rr * WW + cc]);
        }
    bool keep = (p == mx) && (r >= 4) && (r < HH - 4) && (c >= 4) && (c < WW - 4);
    out[(size_t)b * HWV + i] = keep ? p : 0.0f;
}

// ---------------- exact top-k: 64-bit radix select + compaction + bitonic sort ----------------

__global__ void initK(unsigned int* __restrict__ hist, TopkState* __restrict__ st) {
    int i = blockIdx.x * 256 + threadIdx.x;
    if (i < BB * 256) hist[i] = 0u;
    if (i < BB) { st[i].prefix = 0ull; st[i].kr = NK; st[i].cnt = 0u; }
}

__global__ void histPassK(const float* __restrict__ nms, unsigned int* __restrict__ hist,
                          const TopkState* __restrict__ st, int shift, int pass) {
    __shared__ unsigned int h[256];
    int b = blockIdx.y;
    for (int i = threadIdx.x; i < 256; i += blockDim.x) h[i] = 0u;
    __syncthreads();
    unsigned long long pfx = st[b].prefix;
    unsigned long long msk = (pass == 0) ? 0ull : (~0ull << (shift + 8));
    const float* p = nms + (size_t)b * HWV;
    for (int i = blockIdx.x * blockDim.x + threadIdx.x; i < HWV; i += blockDim.x * gridDim.x) {
        unsigned long long key = makeKey(p[i], i);
        if ((key & msk) == (pfx & msk))
            atomicAdd(&h[(unsigned int)(key >> shift) & 255u], 1u);
    }
    __syncthreads();
    for (int i = threadIdx.x; i < 256; i += blockDim.x)
        if (h[i]) atomicAdd(&hist[b * 256 + i], h[i]);
}

__global__ void selectPassK(unsigned int* __restrict__ hist, TopkState* __restrict__ st,
                            int shift) {
    __shared__ unsigned int h[256];
    int b = blockIdx.x;
    h[threadIdx.x] = hist[b * 256 + threadIdx.x];
    __syncthreads();
    if (threadIdx.x == 0) {
        unsigned int k = st[b].kr;
        unsigned int cum = 0;
        int sel = 0;
        for (int bin = 255; bin >= 0; --bin) {
            unsigned int c = h[bin];
            if (cum + c >= k) { sel = bin; st[b].kr = k - cum; break; }
            cum += c;
        }
        st[b].prefix |= ((unsigned long long)sel) << shift;
    }
    __syncthreads();
    hist[b * 256 + threadIdx.x] = 0u;   // ready for next pass
}

__global__ void compactK(const float* __restrict__ nms, TopkState* __restrict__ st,
                         unsigned long long* __restrict__ sel) {
    int b = blockIdx.y;
    unsigned long long thr = st[b].prefix;   // exact NK-th largest key
    const float* p = nms + (size_t)b * HWV;
    for (int i = blockIdx.x * blockDim.x + threadIdx.x; i < HWV; i += blockDim.x * gridDim.x) {
        unsigned long long key = makeKey(p[i], i);
        if (key >= thr) {
            unsigned int pos = atomicAdd(&st[b].cnt, 1u);
            if (pos < NK) sel[(size_t)b * NK + pos] = key;
        }
    }
}

__global__ void __launch_bounds__(1024) sortK(unsigned long long* __restrict__ sel) {
    extern __shared__ unsigned long long s[];   // 8192 * 8 = 64 KB
    int b = blockIdx.x;
    for (int i = threadIdx.x; i < NK; i += blockDim.x) s[i] = sel[(size_t)b * NK + i];
    __syncthreads();
    for (int k = 2; k <= NK; k <<= 1) {
        for (int j = k >> 1; j > 0; j >>= 1) {
            for (int i = threadIdx.x; i < NK; i += blockDim.x) {
                int l = i ^ j;
                if (l > i) {
                    unsigned long long a = s[i], bb = s[l];
                    bool desc = ((i & k) == 0);
                    if (desc ? (a < bb) : (a > bb)) { s[i] = bb; s[l] = a; }
                }
            }
            __syncthreads();
        }
    }
    for (int i = threadIdx.x; i < NK; i += blockDim.x) sel[(size_t)b * NK + i] = s[i];
}

// ---------------- subpixel refinement + outputs ----------------

__global__ void finalizeK(const unsigned long long* __restrict__ sel,
                          const float* __restrict__ sc,
                          float* __restrict__ kps, float* __restrict__ conf) {
    int gid = blockIdx.x * 256 + threadIdx.x;
    if (gid >= BB * NK) return;
    int b = gid / NK, n = gid % NK;
    unsigned long long key = sel[(size_t)b * NK + n];
    float val = __uint_as_float((unsigned int)(key >> 32));
    int idx = (int)(~(unsigned int)(key & 0xffffffffull));
    int r = idx / WW, c = idx % WW;

    float kx = (2.0f * (float)c + 1.0f - (float)WW) / (float)WW;
    float ky = (2.0f * (float)r + 1.0f - (float)HH) / (float)HH;

    const float* sb = sc + (size_t)b * HWV;
    float v[9];
    float m = NEGINF;
#pragma unroll
    for (int ki = 0; ki < 3; ++ki)
#pragma unroll
        for (int kj = 0; kj < 3; ++kj) {
            int rr = r + ki - 1, cc = c + kj - 1;
            float x = (rr >= 0 && rr < HH && cc >= 0 && cc < WW)
                          ? sb[(size_t)rr * WW + cc] : 0.0f;   // zero pad (Unfold)
            v[ki * 3 + kj] = x;
            m = fmaxf(m, x);
        }
    float sum = 0.0f, ox = 0.0f, oy = 0.0f;
#pragma unroll
    for (int t = 0; t < 9; ++t) {
        float e = expf((v[t] - m) * 2.0f);   // temp 0.5
        sum += e;
        ox += e * (float)((t % 3) - 1);
        oy += e * (float)((t / 3) - 1);
    }
    float scx = 2.0f / (float)WW, scy = 2.0f / (float)HH;
    size_t o = (size_t)(b * NK + n);
    kps[o * 2 + 0] = kx + (ox / sum) * scx;
    kps[o * 2 + 1] = ky + (oy / sum) * scy;
    conf[o] = val;
}

// ---------------- host ----------------

extern "C" void kernel_launch(void* const* d_in, const int* in_sizes, int n_in,
                              void* d_out, int out_size, void* d_ws, size_t ws_size,
                              hipStream_t stream) {
    const float* sc = (const float*)d_in[0];
    float* out = (float*)d_out;

    char* ws = (char*)d_ws;
    size_t o = 0;
    auto take = [&](size_t bytes) { size_t r = o; o += (bytes + 255) & ~(size_t)255; return r; };
    float* xh      = (float*)(ws + take((size_t)BB * HWV * 4));   // reused as NMS output
    float* probs   = (float*)(ws + take((size_t)BB * HWV * 4));
    float* partial = (float*)(ws + take((size_t)BB * 256 * 4));
    float* maxv    = (float*)(ws + take((size_t)BB * 4));
    float* invs    = (float*)(ws + take((size_t)BB * 4));
    unsigned int* hist = (unsigned int*)(ws + take((size_t)BB * 256 * 4));
    TopkState* st  = (TopkState*)(ws + take((size_t)BB * sizeof(TopkState)));
    unsigned long long* sel = (unsigned long long*)(ws + take((size_t)BB * NK * 8));

    // softmax statistics
    reduceMaxK<<<dim3(256, BB), 256, 0, stream>>>(sc, partial);
    finalMaxK<<<BB, 256, 0, stream>>>(partial, maxv);
    reduceSumK<<<dim3(256, BB), 256, 0, stream>>>(sc, maxv, partial);
    finalSumK<<<BB, 256, 0, stream>>>(partial, invs);

    // separable coverage conv on WMMA
    int nblocks = (BB * TILES_B) / 4;   // 9604, exact
    convHK<<<nblocks, 128, 0, stream>>>(sc, maxv, invs, xh);
    convVK<<<nblocks, 128, 0, stream>>>(sc, xh, maxv, invs, probs);

    // NMS + border -> reuse xh buffer
    nmsK<<<dim3((HWV + 255) / 256, BB), 256, 0, stream>>>(probs, xh);

    // exact top-NK per batch
    initK<<<(BB * 256 + 255) / 256, 256, 0, stream>>>(hist, st);
    for (int pass = 0; pass < 8; ++pass) {
        int shift = 56 - 8 * pass;
        histPassK<<<dim3(256, BB), 256, 0, stream>>>(xh, hist, st, shift, pass);
        selectPassK<<<BB, 256, 0, stream>>>(hist, st, shift);
    }
    compactK<<<dim3(256, BB), 256, 0, stream>>>(xh, st, sel);
    sortK<<<BB, 1024, NK * 8, stream>>>(sel);

    // outputs: kps [B,NK,2] then conf [B,NK]
    float* kps = out;
    float* conf = out + (size_t)BB * NK * 2;
    finalizeK<<<(BB * NK + 255) / 256, 256, 0, stream>>>(sel, sc, kps, conf);
}